// HGNN_layer_5454608466191
// MI455X (gfx1250) — compile-verified
//
#include <hip/hip_runtime.h>

#define B_ 4
#define N_ 2304
#define C_ 64
#define NT_ 144
#define K1_ 11
#define TKMAX_ 64
#define ELOC_ 484
#define NODE_ 48
#define BN_EPS_ 1e-5f
#define FLTMAX_ 3.402823466e+38f
#define KCHUNK_ 128  // K-depth staged per async chunk in k_aggregate

typedef __attribute__((ext_vector_type(2))) float v2f;
typedef __attribute__((ext_vector_type(8))) float v8f;

__device__ __forceinline__ v8f wmma_f32_16x16x4(v2f a, v2f b, v8f c) {
  // 8-arg form: (neg_a, A, neg_b, B, c_mod, C, reuse_a, reuse_b)
  return __builtin_amdgcn_wmma_f32_16x16x4_f32(false, a, false, b, (short)0, c,
                                               false, false);
}

// ---------------- squared norms per node ----------------
__global__ void k_sqnorm(const float* __restrict__ x, float* __restrict__ sq) {
  int i = blockIdx.x * blockDim.x + threadIdx.x;
  if (i >= B_ * N_) return;
  const float* p = x + (size_t)i * C_;
  float s = 0.f;
#pragma unroll
  for (int c = 0; c < C_; c += 4) {
    float4 v = *(const float4*)(p + c);
    s += v.x * v.x + v.y * v.y + v.z * v.z + v.w * v.w;
  }
  sq[i] = s;
}

// ---------------- pairwise distances via fp32 WMMA ----------------
__global__ void k_dist(const float* __restrict__ x, const float* __restrict__ sq,
                       float* __restrict__ d) {
  int lane = threadIdx.x & 31;
  int wv = threadIdx.x >> 5;
  int ti = blockIdx.x;
  int tj = blockIdx.y * 8 + wv;
  int b = blockIdx.z;
  const float* xb = x + (size_t)b * N_ * C_;
  int mrow = ti * 16 + (lane & 15);
  int ncol = tj * 16 + (lane & 15);
  int khalf = (lane >> 4) << 1;  // lanes 0-15: K=k0,k0+1 ; lanes 16-31: K=k0+2,k0+3
  v8f acc = {};
#pragma unroll
  for (int k0 = 0; k0 < C_; k0 += 4) {
    v2f a = *(const v2f*)(xb + (size_t)mrow * C_ + (k0 + khalf));
    v2f bb = *(const v2f*)(xb + (size_t)ncol * C_ + (k0 + khalf));
    acc = wmma_f32_16x16x4(a, bb, acc);
  }
  float sqn = sq[b * N_ + ncol];
  int mbase = ti * 16 + ((lane >> 4) << 3);
  float* db = d + (size_t)b * N_ * N_;
#pragma unroll
  for (int r = 0; r < 8; ++r) {
    int m = mbase + r;
    db[(size_t)m * N_ + ncol] = sq[b * N_ + m] + sqn - 2.0f * acc[r];
  }
}

// ---------------- iterative top-64 (ascending) per row ----------------
__global__ void k_topk(const float* __restrict__ d, int* __restrict__ inds) {
  int lane = threadIdx.x;
  int n = blockIdx.x, b = blockIdx.y;
  size_t row = (size_t)(b * N_ + n);
  const float* dr = d + row * N_;
  float v[72];  // N_ = 32 * 72
#pragma unroll
  for (int t = 0; t < 72; ++t) v[t] = dr[lane + (t << 5)];
  int* outp = inds + row * TKMAX_;
#pragma unroll 1
  for (int it = 0; it < TKMAX_; ++it) {
    float best = FLTMAX_;
    int bidx = 0x7fffffff;
#pragma unroll
    for (int t = 0; t < 72; ++t) {
      float vv = v[t];
      int c = lane + (t << 5);
      if (vv < best || (vv == best && c < bidx)) { best = vv; bidx = c; }
    }
#pragma unroll
    for (int off = 16; off > 0; off >>= 1) {
      float ov = __shfl_xor(best, off, 32);
      int oi = __shfl_xor(bidx, off, 32);
      if (ov < best || (ov == best && oi < bidx)) { best = ov; bidx = oi; }
    }
    if (lane == 0) outp[it] = bidx;
    bool own = (bidx & 31) == lane;
    int tmark = bidx >> 5;
#pragma unroll
    for (int t = 0; t < 72; ++t)
      if (own && t == tmark) v[t] = FLTMAX_;
  }
}

// ---------------- incoming degree from top-11 ----------------
__global__ void k_degree(const int* __restrict__ inds, int* __restrict__ Dv) {
  int i = blockIdx.x * blockDim.x + threadIdx.x;
  if (i >= B_ * N_) return;
  int b = i / N_;
  const int* p = inds + (size_t)i * TKMAX_;
#pragma unroll
  for (int t = 0; t < K1_; ++t) atomicAdd(&Dv[b * N_ + p[t]], 1);
}

// ---------------- scatter knn hyperedges into M (and DV accumulation) ----
__global__ void k_scatter_knn(const int* __restrict__ inds,
                              const int* __restrict__ Dv, float* __restrict__ G,
                              float* __restrict__ DVK) {
  __shared__ int mem[TKMAX_];
  int lane = threadIdx.x;
  int n = blockIdx.x, b = blockIdx.y;
  int row = b * N_ + n;
  int deg = Dv[row];
  deg = deg < 1 ? 1 : (deg > TKMAX_ ? TKMAX_ : deg);
  const int* p = inds + (size_t)row * TKMAX_;
  for (int t = lane; t < deg; t += 32) mem[t] = p[t];
  __syncthreads();
  float w = 1.0f / (float)deg;  // 1/DE for this edge
  for (int t = lane; t < deg; t += 32) atomicAdd(&DVK[b * N_ + mem[t]], 1.0f);
  float* Gb = G + (size_t)b * N_ * N_;
  for (int i = 0; i < deg; ++i) {
    float* grow = Gb + (size_t)mem[i] * N_;
    for (int j = lane; j < deg; j += 32) atomicAdd(&grow[mem[j]], w);
  }
}

// ---------------- scatter local conv hyperedges (DE = 25) ----------------
__global__ void k_scatter_loc(float* __restrict__ G) {
  __shared__ int mem[25];
  int lane = threadIdx.x;
  int e = blockIdx.x, b = blockIdx.y;
  int py = e / 22, px = e % 22;
  if (lane < 25) mem[lane] = (py * 2 + lane / 5) * NODE_ + (px * 2 + lane % 5);
  __syncthreads();
  const float w = 1.0f / 25.0f;
  float* Gb = G + (size_t)b * N_ * N_;
  for (int i = 0; i < 25; ++i) {
    float* grow = Gb + (size_t)mem[i] * N_;
    if (lane < 25) atomicAdd(&grow[mem[lane]], w);
  }
}

// how many 5-wide stride-2 windows cover coordinate y (0..47)
__device__ __forceinline__ int cov1d(int y) {
  int c = 0;
#pragma unroll
  for (int p = 0; p < 22; ++p) c += (p * 2 <= y && y <= p * 2 + 4) ? 1 : 0;
  return c;
}

__global__ void k_dv2(const float* __restrict__ DVK, float* __restrict__ dv2) {
  int i = blockIdx.x * blockDim.x + threadIdx.x;
  if (i >= B_ * N_) return;
  int n = i % N_;
  float dv = DVK[i] + (float)(cov1d(n / NODE_) * cov1d(n % NODE_));
  dv2[i] = rsqrtf(dv);
}

__global__ void k_norm(float* __restrict__ G, const float* __restrict__ dv2) {
  size_t idx = (size_t)blockIdx.x * blockDim.x + threadIdx.x;
  size_t tot = (size_t)B_ * N_ * N_;
  if (idx >= tot) return;
  int j = (int)(idx % N_);
  size_t bi = idx / N_;
  int i = (int)(bi % N_);
  int b = (int)(bi / N_);
  G[idx] *= dv2[b * N_ + i] * dv2[b * N_ + j];
}

// ---------------- h = x @ W^T + b  via fp32 WMMA ----------------
__global__ void k_linear(const float* __restrict__ x, const float* __restrict__ W,
                         const float* __restrict__ bias, float* __restrict__ h) {
  int lane = threadIdx.x & 31;
  int wv = threadIdx.x >> 5;  // output col tile 0..3
  int ti = blockIdx.x;
  int b = blockIdx.y;
  const float* xb = x + (size_t)b * N_ * C_;
  int mrow = ti * 16 + (lane & 15);
  int ocol = wv * 16 + (lane & 15);
  int khalf = (lane >> 4) << 1;
  v8f acc = {};
#pragma unroll
  for (int k0 = 0; k0 < C_; k0 += 4) {
    v2f a = *(const v2f*)(xb + (size_t)mrow * C_ + (k0 + khalf));
    v2f bb = *(const v2f*)(W + (size_t)ocol * C_ + (k0 + khalf));  // B[k][o]=W[o][k]
    acc = wmma_f32_16x16x4(a, bb, acc);
  }
  float bv = bias[ocol];
  float* hb = h + (size_t)b * N_ * C_;
  int mbase = ti * 16 + ((lane >> 4) << 3);
#pragma unroll
  for (int r = 0; r < 8; ++r)
    hb[(size_t)(mbase + r) * C_ + ocol] = acc[r] + bv;
}

// ---------------- h2 = G @ h  via fp32 WMMA (K = 2304) ----------------
// B-panels (h columns) are double-buffered into LDS with
// global_load_async_to_lds_b128 and synchronized via s_wait_asynccnt.
__global__ void k_aggregate(const float* __restrict__ G,
                            const float* __restrict__ h, float* __restrict__ h2) {
  // per-wave double buffer: KCHUNK_ x 16 floats = 8KB each; 4 waves -> 64KB
  __shared__ float tile[4][2][KCHUNK_ * 16];
  int lane = threadIdx.x & 31;
  int wv = threadIdx.x >> 5;
  int ti = blockIdx.x;
  int b = blockIdx.y;
  const float* Gb = G + (size_t)b * N_ * N_;
  const float* hb = h + (size_t)b * N_ * C_;
  int mrow = ti * 16 + (lane & 15);
  int ocol16 = (lane & 15);
  int khalf = (lane >> 4) << 1;
  const float* grow = Gb + (size_t)mrow * N_;

  // staging role: lane covers row (lane>>2) of the chunk, 4 floats at (lane&3)*4
  const int srow = lane >> 2;
  const int squad = (lane & 3) << 2;
  unsigned stOff = (unsigned)((srow * 16 + squad) * 4);  // byte offset in tile
  unsigned ldsBase[2];
  ldsBase[0] = (unsigned)(size_t)(void*)&tile[wv][0][0];
  ldsBase[1] = (unsigned)(size_t)(void*)&tile[wv][1][0];
  const float* gbase = hb + wv * 16 + squad;

  auto stage = [&](int k0, unsigned base) {
    unsigned dst = base + stOff;
#pragma unroll
    for (int r = 0; r < KCHUNK_; r += 8) {  // 8 rows per b128 across 32 lanes
      unsigned long long ga =
          (unsigned long long)(size_t)(gbase + (size_t)(k0 + srow + r) * C_);
      unsigned da = dst + (unsigned)(r * 16 * 4);
      asm volatile("global_load_async_to_lds_b128 %0, %1, off" ::"v"(da),
                   "v"(ga)
                   : "memory");
    }
  };

  stage(0, ldsBase[0]);
  v8f acc = {};
  const int NCH = N_ / KCHUNK_;
#pragma unroll 1
  for (int ci = 0; ci < NCH; ++ci) {
    int kc = ci * KCHUNK_;
    if (ci + 1 < NCH) {
      stage(kc + KCHUNK_, ldsBase[(ci + 1) & 1]);
      // newest 16 async loads (next chunk) may stay outstanding;
      // everything older (current chunk) must have landed in LDS
      asm volatile("s_wait_asynccnt 0x10" ::: "memory");
    } else {
      asm volatile("s_wait_asynccnt 0x0" ::: "memory");
    }
    const float* cur = &tile[wv][ci & 1][0];
    __builtin_prefetch(grow + kc + KCHUNK_, 0, 1);
#pragma unroll 8
    for (int kk = 0; kk < KCHUNK_; kk += 4) {
      v2f a = *(const v2f*)(grow + kc + kk + khalf);
      v2f bb;
      bb[0] = cur[(kk + khalf) * 16 + ocol16];
      bb[1] = cur[(kk + khalf + 1) * 16 + ocol16];
      acc = wmma_f32_16x16x4(a, bb, acc);
    }
  }

  float* ob = h2 + (size_t)b * N_ * C_;
  int ocol = wv * 16 + ocol16;
  int mbase = ti * 16 + ((lane >> 4) << 3);
#pragma unroll
  for (int r = 0; r < 8; ++r)
    ob[(size_t)(mbase + r) * C_ + ocol] = acc[r];
}

// ---------------- batchnorm stats (sum, sumsq per channel) ----------------
__global__ void k_bnstats(const float* __restrict__ h2, float* __restrict__ stat) {
  __shared__ float ssum[256], ssq[256];
  int t = threadIdx.x;
  int c = t & 63, rl = t >> 6;
  int r0 = blockIdx.x * 256;
  float s = 0.f, q = 0.f;
  for (int r = r0 + rl; r < r0 + 256; r += 4) {
    float v = h2[(size_t)r * C_ + c];
    s += v;
    q += v * v;
  }
  ssum[t] = s;
  ssq[t] = q;
  __syncthreads();
  if (t < 64) {
    float ts = ssum[t] + ssum[t + 64] + ssum[t + 128] + ssum[t + 192];
    float tq = ssq[t] + ssq[t + 64] + ssq[t + 128] + ssq[t + 192];
    atomicAdd(&stat[c], ts);
    atomicAdd(&stat[64 + c], tq);
  }
}

// ---------------- BN + ReLU + residual ----------------
__global__ void k_final(const float* __restrict__ h2, const float* __restrict__ x,
                        const float* __restrict__ gamma,
                        const float* __restrict__ beta,
                        const float* __restrict__ stat, float* __restrict__ out) {
  int idx = blockIdx.x * blockDim.x + threadIdx.x;
  if (idx >= B_ * N_ * C_) return;
  int c = idx & 63;
  const float inv = 1.0f / (float)(B_ * N_);
  float mean = stat[c] * inv;
  float var = stat[64 + c] * inv - mean * mean;
  float yv = gamma[c] * (h2[idx] - mean) * rsqrtf(var + BN_EPS_) + beta[c];
  out[idx] = fmaxf(yv, 0.0f) + x[idx];
}

extern "C" void kernel_launch(void* const* d_in, const int* in_sizes, int n_in,
                              void* d_out, int out_size, void* d_ws,
                              size_t ws_size, hipStream_t stream) {
  const float* x = (const float*)d_in[0];
  const float* W = (const float*)d_in[1];
  const float* bias = (const float*)d_in[2];
  const float* gamma = (const float*)d_in[3];
  const float* beta = (const float*)d_in[4];
  float* out = (float*)d_out;

  // workspace layout (floats)
  float* ws = (float*)d_ws;
  const size_t NN = (size_t)N_ * N_;
  float* dD = ws;                         // B*N*N  distances
  float* G = dD + (size_t)B_ * NN;        // B*N*N  graph
  float* h = G + (size_t)B_ * NN;         // B*N*C
  float* h2 = h + (size_t)B_ * N_ * C_;   // B*N*C
  float* sq = h2 + (size_t)B_ * N_ * C_;  // B*N
  float* dv2 = sq + B_ * N_;              // B*N
  float* DVK = dv2 + B_ * N_;             // B*N
  float* stat = DVK + B_ * N_;            // 128
  int* inds = (int*)(stat + 128);         // B*N*64
  int* Dv = inds + (size_t)B_ * N_ * TKMAX_;  // B*N

  hipMemsetAsync(G, 0, (size_t)B_ * NN * sizeof(float), stream);
  hipMemsetAsync(DVK, 0, (size_t)B_ * N_ * sizeof(float), stream);
  hipMemsetAsync(stat, 0, 128 * sizeof(float), stream);
  hipMemsetAsync(Dv, 0, (size_t)B_ * N_ * sizeof(int), stream);

  k_sqnorm<<<dim3((B_ * N_ + 255) / 256), 256, 0, stream>>>(x, sq);
  k_dist<<<dim3(NT_, NT_ / 8, B_), 256, 0, stream>>>(x, sq, dD);
  k_topk<<<dim3(N_, B_), 32, 0, stream>>>(dD, inds);
  k_degree<<<dim3((B_ * N_ + 255) / 256), 256, 0, stream>>>(inds, Dv);
  k_scatter_knn<<<dim3(N_, B_), 32, 0, stream>>>(inds, Dv, G, DVK);
  k_scatter_loc<<<dim3(ELOC_, B_), 32, 0, stream>>>(G);
  k_dv2<<<dim3((B_ * N_ + 255) / 256), 256, 0, stream>>>(DVK, dv2);
  size_t tot = (size_t)B_ * NN;
  k_norm<<<dim3((unsigned)((tot + 255) / 256)), 256, 0, stream>>>(G, dv2);
  k_linear<<<dim3(NT_, B_), 128, 0, stream>>>(x, W, bias, h);
  k_aggregate<<<dim3(NT_, B_), 128, 0, stream>>>(G, h, h2);
  k_bnstats<<<dim3((B_ * N_) / 256), 256, 0, stream>>>(h2, stat);
  k_final<<<dim3((B_ * N_ * C_ + 255) / 256), 256, 0, stream>>>(h2, x, gamma,
                                                               beta, stat, out);
}